// nvs_no_skip_27745488732698
// MI455X (gfx1250) — compile-verified
//
#include <hip/hip_runtime.h>
#include <math.h>

#define TPB 256

typedef float v2f __attribute__((ext_vector_type(2)));
typedef float v8f __attribute__((ext_vector_type(8)));

static inline int gridFor(long long n) {
    long long g = (n + TPB - 1) / TPB;
    if (g > 262144) g = 262144;
    if (g < 1) g = 1;
    return (int)g;
}
static inline int cdiv_i(long long a, long long b) { return (int)((a + b - 1) / b); }
static inline int ilog2_i(int v) { int s = 0; while ((1 << s) < v) ++s; return s; }

// ---------------------------------------------------------------- fill
__global__ void nvs_fill_f32(float* p, long long n, float v) {
    long long i = (long long)blockIdx.x * blockDim.x + threadIdx.x;
    long long st = (long long)gridDim.x * blockDim.x;
    for (; i < n; i += st) p[i] = v;
}
__global__ void nvs_fill_i32(int* p, long long n, int v) {
    long long i = (long long)blockIdx.x * blockDim.x + threadIdx.x;
    long long st = (long long)gridDim.x * blockDim.x;
    for (; i < n; i += st) p[i] = v;
}

// ---------------------------------------------------------------- degree
__global__ void nvs_deg(const int* __restrict__ ei, int E, float* __restrict__ deg) {
    long long i = (long long)blockIdx.x * blockDim.x + threadIdx.x;
    long long st = (long long)gridDim.x * blockDim.x;
    for (; i < E; i += st) atomicAdd(&deg[ei[E + i]], 1.0f);
}

// ---------------------------------------------------------------- spline scatter
// s is chunk-local: row = dst - c0, layout [rows, 27*Fin]. Fin = 1<<fshift.
__global__ void nvs_scatter(const int* __restrict__ ei, int E,
                            const float* __restrict__ ea,
                            const float* __restrict__ x, int fshift,
                            float* __restrict__ s, int c0, int rows) {
    const int Fin = 1 << fshift;
    const int fmask = Fin - 1;
    long long total = (long long)E << fshift;
    long long i = (long long)blockIdx.x * blockDim.x + threadIdx.x;
    long long st = (long long)gridDim.x * blockDim.x;
    for (; i < total; i += st) {
        int e = (int)(i >> fshift);
        int f = (int)i & fmask;
        int dst = ei[E + e];
        int r = dst - c0;
        if (r < 0 || r >= rows) continue;
        int src = ei[e];
        float xv = x[((long long)src << fshift) + f];
        float fr[3]; int k0[3];
#pragma unroll
        for (int d = 0; d < 3; ++d) {
            float p = ea[(long long)e * 3 + d];
            p = fminf(fmaxf(p, 0.0f), 1.0f) * 2.0f;      // * (K-1)
            float fl = fminf(floorf(p), 1.0f);            // clip to K-2
            k0[d] = (int)fl;
            fr[d] = p - fl;
        }
        float* srow = s + ((long long)r * 27 << fshift) + f;
#pragma unroll
        for (int b = 0; b < 8; ++b) {
            int bx = b & 1, by = (b >> 1) & 1, bz = (b >> 2) & 1;
            float w = (bx ? fr[0] : 1.0f - fr[0]) *
                      (by ? fr[1] : 1.0f - fr[1]) *
                      (bz ? fr[2] : 1.0f - fr[2]);
            int wi = ((k0[0] + bx) * 3 + (k0[1] + by)) * 3 + (k0[2] + bz);
            atomicAdd(srow + ((long long)wi << fshift), w * xv);
        }
    }
}

// scale row r of s by 1/max(deg[c0+r],1): one block per row, no divides
__global__ void nvs_scale_rows(float* __restrict__ s, int rowlen,
                               const float* __restrict__ deg, int c0) {
    const int r = blockIdx.x;
    const float inv = 1.0f / fmaxf(deg[c0 + r], 1.0f);
    float* row = s + (long long)r * rowlen;
    for (int i = threadIdx.x; i < rowlen; i += blockDim.x) row[i] *= inv;
}

// ---------------------------------------------------------------- fp32 WMMA GEMM
// C[M,N] = A1[M,K1]*B1[K1,N] + A2[M,K2]*B2[K2,N] + bias ; optional ELU.
// One wave32 per block computes a 16x64 C tile: 4 16x16 WMMA accumulators
// sharing each A fragment (V_WMMA_F32_16X16X4_F32).
// B addressing: wave-uniform scalar base (B + k*N, SGPR, bumped per k-step)
// + loop-invariant lane offset (half*2*N + col) -> saddr+voffset loads,
// no per-load 64-bit VALU address math.
// Out-of-range rows/cols are CLAMPED (loads always legal); stores guarded.
__global__ void __launch_bounds__(32)
nvs_gemm_wmma_f32(const float* __restrict__ A1, int K1, const float* __restrict__ B1,
                  const float* __restrict__ A2, int K2, const float* __restrict__ B2,
                  const float* __restrict__ bias, float* __restrict__ C,
                  int M, int N, int do_elu) {
    const int lane = threadIdx.x & 31;
    const int half = lane >> 4;
    const int l    = lane & 15;
    const int m0 = blockIdx.x * 16;
    const int n0 = blockIdx.y * 64;

    const int arow  = m0 + l;
    const int arowc = (arow < M) ? arow : (M - 1);

    int colv[4], colc[4], offc[4];
#pragma unroll
    for (int t = 0; t < 4; ++t) {
        colv[t] = n0 + t * 16 + l;
        colc[t] = (colv[t] < N) ? colv[t] : (N - 1);
        offc[t] = half * 2 * N + colc[t];   // lane-constant across k
    }

    v8f acc[4] = {v8f{}, v8f{}, v8f{}, v8f{}};

#pragma unroll 1
    for (int pass = 0; pass < 2; ++pass) {
        const float* __restrict__ A = pass ? A2 : A1;
        const float* __restrict__ B = pass ? B2 : B1;
        const int K = pass ? K2 : K1;
        const float* pA  = A + (long long)arowc * K + half * 2;  // per-lane base
        const float* pBk = B;                                    // wave-uniform base
        int k = 0;
        for (; k + 3 < K; k += 4) {
            v2f a;
            a.x = pA[0];
            a.y = pA[1];
            __builtin_prefetch(pA + 32, 0, 0);
#pragma unroll
            for (int t = 0; t < 4; ++t) {
                v2f b;
                b.x = pBk[offc[t]];
                b.y = pBk[offc[t] + N];
                acc[t] = __builtin_amdgcn_wmma_f32_16x16x4_f32(
                             false, a, false, b, (short)0, acc[t], false, false);
            }
            pA += 4;
            pBk += (long long)4 * N;
        }
        if (k < K) {  // K % 4 tail: unconditional clamped loads + value selects
            const int ka = k + half * 2;
            const int i0 = (ka     < K) ? ka     : (K - 1);
            const int i1 = (ka + 1 < K) ? ka + 1 : (K - 1);
            const float* Ab = A + (long long)arowc * K;
            float va0 = Ab[i0];
            float va1 = Ab[i1];
            v2f a;
            a.x = (ka     < K) ? va0 : 0.0f;
            a.y = (ka + 1 < K) ? va1 : 0.0f;
#pragma unroll
            for (int t = 0; t < 4; ++t) {
                float vb0 = B[(long long)i0 * N + colc[t]];
                float vb1 = B[(long long)i1 * N + colc[t]];
                v2f b;
                b.x = (ka     < K) ? vb0 : 0.0f;
                b.y = (ka + 1 < K) ? vb1 : 0.0f;
                acc[t] = __builtin_amdgcn_wmma_f32_16x16x4_f32(
                             false, a, false, b, (short)0, acc[t], false, false);
            }
        }
    }

    // epilogue: bias + optional ELU, guarded stores
#pragma unroll 1
    for (int t = 0; t < 4; ++t) {
        const int col = colv[t];
        if (col < N) {
            const float bv = bias[col];
#pragma unroll
            for (int j = 0; j < 8; ++j) {
                int row = m0 + j + half * 8;
                if (row < M) {
                    float v = acc[t][j] + bv;
                    if (do_elu) v = (v > 0.0f) ? v : (expf(v) - 1.0f);
                    C[(long long)row * N + col] = v;
                }
            }
        }
    }
}

// ---------------------------------------------------------------- batch norm
__global__ void nvs_bn_stats(const float* __restrict__ x, int M, int C,
                             float* __restrict__ mean, float* __restrict__ var) {
    const int c = blockIdx.x;
    __shared__ float ssum[TPB], ssq[TPB];
    float s = 0.0f, q = 0.0f;
    for (int r = threadIdx.x; r < M; r += blockDim.x) {
        float v = x[(long long)r * C + c];
        s += v; q += v * v;
    }
    ssum[threadIdx.x] = s; ssq[threadIdx.x] = q;
    __syncthreads();
    for (int st = TPB / 2; st > 0; st >>= 1) {
        if (threadIdx.x < st) {
            ssum[threadIdx.x] += ssum[threadIdx.x + st];
            ssq[threadIdx.x]  += ssq[threadIdx.x + st];
        }
        __syncthreads();
    }
    if (threadIdx.x == 0) {
        float mu = ssum[0] / (float)M;
        mean[c] = mu;
        var[c]  = ssq[0] / (float)M - mu * mu;
    }
}
// C = 1<<cshift
__global__ void nvs_bn_apply(float* __restrict__ x, long long n, int cshift,
                             const float* __restrict__ mean, const float* __restrict__ var,
                             const float* __restrict__ gamma, const float* __restrict__ beta) {
    const int cmask = (1 << cshift) - 1;
    long long i = (long long)blockIdx.x * blockDim.x + threadIdx.x;
    long long st = (long long)gridDim.x * blockDim.x;
    for (; i < n; i += st) {
        int c = (int)i & cmask;
        x[i] = gamma[c] * (x[i] - mean[c]) * rsqrtf(var[c] + 1e-5f) + beta[c];
    }
}

// ---------------------------------------------------------------- segment-max pool
__device__ inline void nvs_atomic_max_f(float* a, float v) {
    if (v >= 0.0f) atomicMax((int*)a, __float_as_int(v));
    else           atomicMin((unsigned int*)a, __float_as_uint(v));
}
__global__ void nvs_seg_cnt(const int* __restrict__ cl, int N, int* __restrict__ cnt) {
    long long i = (long long)blockIdx.x * blockDim.x + threadIdx.x;
    long long st = (long long)gridDim.x * blockDim.x;
    for (; i < N; i += st) atomicAdd(&cnt[cl[i]], 1);
}
__global__ void nvs_pool_max(const float* __restrict__ x, long long n, int cshift,
                             const int* __restrict__ cl, float* __restrict__ out) {
    const int cmask = (1 << cshift) - 1;
    long long i = (long long)blockIdx.x * blockDim.x + threadIdx.x;
    long long st = (long long)gridDim.x * blockDim.x;
    for (; i < n; i += st) {
        int node = (int)(i >> cshift);
        int c = (int)i & cmask;
        nvs_atomic_max_f(&out[((long long)cl[node] << cshift) + c], x[i]);
    }
}
__global__ void nvs_pool_fin(float* __restrict__ out, long long n, int cshift,
                             const int* __restrict__ cnt) {
    long long i = (long long)blockIdx.x * blockDim.x + threadIdx.x;
    long long st = (long long)gridDim.x * blockDim.x;
    for (; i < n; i += st)
        if (cnt[i >> cshift] == 0) out[i] = 0.0f;
}

// ---------------------------------------------------------------- host helpers
static void nvs_run_layer(const float* hin, int N, int Fin, int Fout,
                          const int* ei, int E, const float* ea,
                          const float* W, const float* root, const float* bias,
                          const float* gamma, const float* beta,
                          float* hout, float* sbuf, int nchunk,
                          float* deg, float* meanB, float* varB, hipStream_t stream) {
    nvs_fill_f32<<<gridFor(N), TPB, 0, stream>>>(deg, N, 0.0f);
    nvs_deg<<<gridFor(E), TPB, 0, stream>>>(ei, E, deg);
    const int fshift = ilog2_i(Fin);
    const int KDF = 27 * Fin;
    for (int c0 = 0; c0 < N; c0 += nchunk) {
        int rows = (N - c0 < nchunk) ? (N - c0) : nchunk;
        long long sElems = (long long)rows * KDF;
        nvs_fill_f32<<<gridFor(sElems), TPB, 0, stream>>>(sbuf, sElems, 0.0f);
        nvs_scatter<<<gridFor((long long)E * Fin), TPB, 0, stream>>>(
            ei, E, ea, hin, fshift, sbuf, c0, rows);
        nvs_scale_rows<<<rows, TPB, 0, stream>>>(sbuf, KDF, deg, c0);
        dim3 g(cdiv_i(rows, 16), cdiv_i(Fout, 64));
        nvs_gemm_wmma_f32<<<g, 32, 0, stream>>>(
            sbuf, KDF, W, hin + (long long)c0 * Fin, Fin, root, bias,
            hout + (long long)c0 * Fout, rows, Fout, 1);
    }
    nvs_bn_stats<<<Fout, TPB, 0, stream>>>(hout, N, Fout, meanB, varB);
    nvs_bn_apply<<<gridFor((long long)N * Fout), TPB, 0, stream>>>(
        hout, (long long)N * Fout, ilog2_i(Fout), meanB, varB, gamma, beta);
}

static void nvs_run_pool(const float* x, int N, int C, const int* cl, int Nseg,
                         float* out, int* cnt, hipStream_t stream) {
    long long n = (long long)Nseg * C;
    const int cshift = ilog2_i(C);
    nvs_fill_f32<<<gridFor(n), TPB, 0, stream>>>(out, n, -INFINITY);
    nvs_fill_i32<<<gridFor(Nseg), TPB, 0, stream>>>(cnt, Nseg, 0);
    nvs_seg_cnt<<<gridFor(N), TPB, 0, stream>>>(cl, N, cnt);
    nvs_pool_max<<<gridFor((long long)N * C), TPB, 0, stream>>>(x, (long long)N * C, cshift, cl, out);
    nvs_pool_fin<<<gridFor(n), TPB, 0, stream>>>(out, n, cshift, cnt);
}

extern "C" void kernel_launch(void* const* d_in, const int* in_sizes, int n_in,
                              void* d_out, int out_size, void* d_ws, size_t ws_size,
                              hipStream_t stream) {
    // setup_inputs() dict order
    const float* x   = (const float*)d_in[0];
    const int*   ei1 = (const int*)d_in[1];
    const float* ea1 = (const float*)d_in[2];
    const int*   cl1 = (const int*)d_in[3];
    const int*   ei2 = (const int*)d_in[4];
    const float* ea2 = (const float*)d_in[5];
    const int*   cl2 = (const int*)d_in[6];
    const int*   ei3 = (const int*)d_in[7];
    const float* ea3 = (const float*)d_in[8];
    const int*   cl3 = (const int*)d_in[9];
    const int*   ei4 = (const int*)d_in[10];
    const float* ea4 = (const float*)d_in[11];
    const int*   cl4 = (const int*)d_in[12];
    const float* Wl[4]    = {(const float*)d_in[13], (const float*)d_in[18], (const float*)d_in[23], (const float*)d_in[28]};
    const float* rootl[4] = {(const float*)d_in[14], (const float*)d_in[19], (const float*)d_in[24], (const float*)d_in[29]};
    const float* biasl[4] = {(const float*)d_in[15], (const float*)d_in[20], (const float*)d_in[25], (const float*)d_in[30]};
    const float* gaml[4]  = {(const float*)d_in[16], (const float*)d_in[21], (const float*)d_in[26], (const float*)d_in[31]};
    const float* betl[4]  = {(const float*)d_in[17], (const float*)d_in[22], (const float*)d_in[27], (const float*)d_in[32]};
    const float* fc1w = (const float*)d_in[33];
    const float* fc1b = (const float*)d_in[34];
    const float* fc2w = (const float*)d_in[35];
    const float* fc2b = (const float*)d_in[36];

    const int N1 = in_sizes[0],  E1 = in_sizes[1] / 2;
    const int N2 = in_sizes[6],  E2 = in_sizes[4] / 2;
    const int N3 = in_sizes[9],  E3 = in_sizes[7] / 2;
    const int N4 = in_sizes[12], E4 = in_sizes[10] / 2;
    const int NP = 8 * 64; // NUM_GRAPHS * POOL_SIZE

    // ---- carve workspace
    size_t off = 0;
    char* base = (char*)d_ws;
    auto carve = [&](size_t bytes) -> char* {
        char* p = base + off;
        off += (bytes + 255) & ~(size_t)255;
        return p;
    };
    auto maxs = [](size_t a, size_t b) { return a > b ? a : b; };
    size_t convMax = (size_t)N1 * 64;
    convMax = maxs(convMax, (size_t)N2 * 128);
    convMax = maxs(convMax, (size_t)N3 * 256);
    convMax = maxs(convMax, (size_t)N4 * 512);
    convMax = maxs(convMax, (size_t)8 * 1024);
    size_t poolMax = (size_t)N2 * 64;
    poolMax = maxs(poolMax, (size_t)N3 * 128);
    poolMax = maxs(poolMax, (size_t)N4 * 256);
    poolMax = maxs(poolMax, (size_t)NP * 512);

    float* bufA  = (float*)carve(convMax * 4);
    float* bufB  = (float*)carve(poolMax * 4);
    float* deg   = (float*)carve((size_t)N1 * 4);
    int segMax = N2 > NP ? N2 : NP;
    if (N3 > segMax) segMax = N3;
    if (N4 > segMax) segMax = N4;
    int*   cnt   = (int*)carve((size_t)segMax * 4);
    float* meanB = (float*)carve(1024 * 4);
    float* varB  = (float*)carve(1024 * 4);
    size_t rem = (ws_size > off) ? (ws_size - off) : 0;
    long long nchunk = (long long)(rem / ((size_t)27 * 256 * 4));
    if (nchunk > 2048) nchunk = 2048;
    if (nchunk < 64)   nchunk = 64;
    float* sbuf = (float*)carve((size_t)nchunk * 27 * 256 * 4);

    // ---- layer 1: (N1,1) -> (N1,64) -> pool -> (N2,64)
    nvs_run_layer(x, N1, 1, 64, ei1, E1, ea1, Wl[0], rootl[0], biasl[0], gaml[0], betl[0],
                  bufA, sbuf, (int)nchunk, deg, meanB, varB, stream);
    nvs_run_pool(bufA, N1, 64, cl1, N2, bufB, cnt, stream);

    // ---- layer 2: (N2,64) -> (N2,128) -> pool -> (N3,128)
    nvs_run_layer(bufB, N2, 64, 128, ei2, E2, ea2, Wl[1], rootl[1], biasl[1], gaml[1], betl[1],
                  bufA, sbuf, (int)nchunk, deg, meanB, varB, stream);
    nvs_run_pool(bufA, N2, 128, cl2, N3, bufB, cnt, stream);

    // ---- layer 3: (N3,128) -> (N3,256) -> pool -> (N4,256)
    nvs_run_layer(bufB, N3, 128, 256, ei3, E3, ea3, Wl[2], rootl[2], biasl[2], gaml[2], betl[2],
                  bufA, sbuf, (int)nchunk, deg, meanB, varB, stream);
    nvs_run_pool(bufA, N3, 256, cl3, N4, bufB, cnt, stream);

    // ---- layer 4: (N4,256) -> (N4,512) -> pool -> (512,512)
    nvs_run_layer(bufB, N4, 256, 512, ei4, E4, ea4, Wl[3], rootl[3], biasl[3], gaml[3], betl[3],
                  bufA, sbuf, (int)nchunk, deg, meanB, varB, stream);
    nvs_run_pool(bufA, N4, 512, cl4, NP, bufB, cnt, stream);

    // ---- fc1: (8, 32768) @ (32768, 1024) + b, ELU     (bufB viewed as 8x32768)
    {
        dim3 g(cdiv_i(8, 16), cdiv_i(1024, 64));
        nvs_gemm_wmma_f32<<<g, 32, 0, stream>>>(
            bufB, 64 * 512, fc1w, bufB, 0, fc1w, fc1b, bufA, 8, 1024, 1);
    }
    // ---- fc2: (8, 1024) @ (1024, 2) + b
    {
        dim3 g(cdiv_i(8, 16), cdiv_i(2, 64));
        nvs_gemm_wmma_f32<<<g, 32, 0, stream>>>(
            bufA, 1024, fc2w, bufA, 0, fc2w, fc2b, (float*)d_out, 8, 2, 0);
    }
}